// SymmetricContraction_7645041787178
// MI455X (gfx1250) — compile-verified
//
#include <hip/hip_runtime.h>

// MI455X / gfx1250, wave32. Symmetric contraction as a shared-U GEMM:
//   rows = (b,c) pairs (262144), K = padded monomial index (1092), N = 48 (33 used).
// Dominant compute on V_WMMA_F32_16X16X4_F32; B-matrix staged block-wide in LDS in a
// K-pair-interleaved layout so each B fragment is a single ds_load_b64; branchless
// A-fragments; fused per-element weight epilogue.

typedef __attribute__((ext_vector_type(2))) float v2f;
typedef __attribute__((ext_vector_type(8))) float v8f;

#define NCOLP 48              // padded column count (>= 33 actual columns)
#define KPAD  1092            // 91 groups x 12 K-slots
#define KB2   972             // start of degree-2 K rows (group 81)
#define KB1   1080            // start of degree-1 K rows (group 90)
#define GROUPS    91
#define CH_GROUPS 7           // groups per LDS chunk
#define CHUNKS    13          // 13 * 7 == 91
#define CH_ROWS   (CH_GROUPS * 12)        // 84
#define CH_FLOATS (CH_ROWS * NCOLP)       // 4032 floats = 16.1 KB

// ---------------------------------------------------------------------------
// Kernel 1: pack U1/U2/U3 of both blocks into zero-padded [KPAD][NCOLP],
// stored K-row-pair interleaved:  element (k, n) -> [(k/2)*NCOLP + n]*2 + (k&1).
// A wmma B-fragment (rows k,k+1 at column n) is then one contiguous b64.
// K layout: group g = (i,j) pair (g<81) / i (81<=g<90) / unit (g==90),
// 12 slots each, slot s<9 valid.
// Column layout: [b0:nu3 P3_0][b0:nu2 P2_0][b0:nu1 P1_0]
//                [b1:nu3 3*P3_1][b1:nu2 3*P2_1][b1:nu1 3*P1_1]  (d-major)
// ---------------------------------------------------------------------------
__global__ void pack_ucat_kernel(
    const float* __restrict__ U1_0, const float* __restrict__ U2_0, const float* __restrict__ U3_0,
    const float* __restrict__ U1_1, const float* __restrict__ U2_1, const float* __restrict__ U3_1,
    float* __restrict__ Ucat,
    int P1_0, int P2_0, int P3_0, int P1_1, int P2_1, int P3_1)
{
    int idx = blockIdx.x * blockDim.x + threadIdx.x;
    if (idx >= KPAD * NCOLP) return;
    int k = idx / NCOLP;
    int n = idx % NCOLP;

    int knu = 0, mIdx = 0;
    if (k < KB2)      { int pr = k / 12, s = k % 12; if (s < 9) { knu = 3; mIdx = pr * 9 + s; } }
    else if (k < KB1) { int q = k - KB2; int i = q / 12, s = q % 12; if (s < 9) { knu = 2; mIdx = i * 9 + s; } }
    else              { int s = k - KB1; if (s < 9) { knu = 1; mIdx = s; } }

    float val = 0.0f;
    if (knu) {
        int c0 = P3_0, c1 = c0 + P2_0, c2 = c1 + P1_0;
        int c3 = c2 + 3 * P3_1, c4 = c3 + 3 * P2_1, c5 = c4 + 3 * P1_1;
        if (n < c0)      { if (knu == 3) val = U3_0[mIdx * P3_0 + n]; }
        else if (n < c1) { if (knu == 2) val = U2_0[mIdx * P2_0 + (n - c0)]; }
        else if (n < c2) { if (knu == 1) val = U1_0[mIdx * P1_0 + (n - c1)]; }
        else if (n < c3) { if (knu == 3) { int t = n - c2; int d = t / P3_1, p = t % P3_1;
                                           val = U3_1[(d * 729 + mIdx) * P3_1 + p]; } }
        else if (n < c4) { if (knu == 2) { int t = n - c3; int d = t / P2_1, p = t % P2_1;
                                           val = U2_1[(d * 81 + mIdx) * P2_1 + p]; } }
        else if (n < c5) { if (knu == 1) { int t = n - c4; int d = t / P1_1, p = t % P1_1;
                                           val = U1_1[(d * 9 + mIdx) * P1_1 + p]; } }
    }
    // pair-interleaved store
    Ucat[((k >> 1) * NCOLP + n) * 2 + (k & 1)] = val;
}

// ---------------------------------------------------------------------------
// Kernel 2: per-wave 16-row GEMM tile via WMMA f32 16x16x4 + fused epilogue.
// 8 waves/block; B-matrix chunks staged cooperatively in LDS.
// ---------------------------------------------------------------------------
__global__ __launch_bounds__(256) void symcon_kernel(
    const float* __restrict__ x, const float* __restrict__ y,
    const float* __restrict__ Ucat,
    const float* __restrict__ W1_0, const float* __restrict__ W2_0, const float* __restrict__ W3_0,
    const float* __restrict__ W1_1, const float* __restrict__ W2_1, const float* __restrict__ W3_1,
    float* __restrict__ out,
    int P1_0, int P2_0, int P3_0, int P1_1, int P2_1, int P3_1)
{
    __shared__ float xs[8][16 * 9];                  // per-wave x rows (dynamic i/j reads)
    __shared__ float qs[8][16 * NCOLP];              // per-wave Q tile for epilogue
    __shared__ __align__(16) float ub[CH_FLOATS];    // block-shared Ucat chunk (pair-interleaved)

    const int lane = threadIdx.x & 31;
    const int wave = threadIdx.x >> 5;
    const int tile = blockIdx.x * 8 + wave;
    const int r16  = lane & 15;
    const bool hi  = lane >= 16;

    // Per-lane register copy of this row's 9 x values (lanes 16-31 duplicate row r16).
    const int rowg = tile * 16 + r16;
    float xv[9];
    #pragma unroll
    for (int e = 0; e < 9; ++e) xv[e] = x[rowg * 9 + e];

    // Stage x rows to LDS once (for uniform-index i/j multiplier reads).
    if (!hi) {
        #pragma unroll
        for (int e = 0; e < 9; ++e) xs[wave][r16 * 9 + e] = xv[e];
    }

    // Pre-masked, pre-selected k-dimension operands: step s covers K-slots 4s..4s+3.
    // This lane contributes slots (4s + 2*hi, +1); slot >= 9 is zero padding.
    float xk0[3], xk1[3];
    xk0[0] = hi ? xv[2] : xv[0];   xk1[0] = hi ? xv[3] : xv[1];
    xk0[1] = hi ? xv[6] : xv[4];   xk1[1] = hi ? xv[7] : xv[5];
    xk0[2] = hi ? 0.0f  : xv[8];   xk1[2] = 0.0f;

    const v8f vzero = {0.f,0.f,0.f,0.f,0.f,0.f,0.f,0.f};
    v8f acc[3];
    #pragma unroll
    for (int t = 0; t < 3; ++t) acc[t] = vzero;

    const float* xrow = &xs[wave][r16 * 9];
    const int kp_hi = hi ? 1 : 0;    // K-pair select within a 4-slot step

    for (int ch = 0; ch < CHUNKS; ++ch) {
        __syncthreads();   // previous chunk fully consumed (also covers xs init)

        // Cooperative b128 stage of this chunk; prefetch next chunk into L2.
        {
            const float4* src = (const float4*)(Ucat + ch * CH_FLOATS);
            float4*       dst = (float4*)ub;
            for (int t = threadIdx.x; t < CH_FLOATS / 4; t += 256) dst[t] = src[t];
            if (ch + 1 < CHUNKS)
                __builtin_prefetch((const char*)(Ucat + (ch + 1) * CH_FLOATS) + threadIdx.x * 64, 0, 0);
        }
        __syncthreads();

        #pragma unroll
        for (int gg = 0; gg < CH_GROUPS; ++gg) {
            const int g = ch * CH_GROUPS + gg;   // uniform across block
            float m;
            if (g < 81)      m = xrow[g / 9] * xrow[g % 9];   // degree-3: x_i * x_j
            else if (g < 90) m = xrow[g - 81];                // degree-2: x_i
            else             m = 1.0f;                        // degree-1

            const int kpbase = gg * 6;           // K-pairs per group = 12/2
            #pragma unroll
            for (int s = 0; s < 3; ++s) {
                v2f A = { m * xk0[s], m * xk1[s] };
                const int kp = kpbase + 2 * s + kp_hi;
                const v2f* bp = (const v2f*)&ub[(kp * NCOLP + r16) * 2];
                #pragma unroll
                for (int t = 0; t < 3; ++t) {
                    v2f B = bp[t * 16];          // one ds_load_b64: rows (2kp, 2kp+1) at col n
                    acc[t] = __builtin_amdgcn_wmma_f32_16x16x4_f32(
                                 false, A, false, B, (short)0, acc[t], false, false);
                }
            }
        }
    }

    // Spill Q tile to LDS. C/D layout: VGPR r = row r (lanes 0-15) / r+8 (16-31),
    // column = lane%16 + 16*t.
    const int Mbase = hi ? 8 : 0;
    #pragma unroll
    for (int t = 0; t < 3; ++t) {
        #pragma unroll
        for (int rr = 0; rr < 8; ++rr)
            qs[wave][(Mbase + rr) * NCOLP + t * 16 + r16] = acc[t][rr];
    }
    __syncthreads();

    // Fused epilogue: apply per-element weights, write output.
    if (!hi) {
        int b = tile >> 3;                     // 8 tiles per batch (C=128)
        int c = (tile & 7) * 16 + r16;         // channel
        int e = 0;
        for (int t = 0; t < 10; ++t) if (y[b * 10 + t] > 0.5f) e = t;

        const float* q = &qs[wave][r16 * NCOLP];
        float o0 = 0.f;
        float o1[3] = {0.f, 0.f, 0.f};
        int n = 0;
        for (int p = 0; p < P3_0; ++p) o0 += q[n++] * W3_0[(e * P3_0 + p) * 128 + c];
        for (int p = 0; p < P2_0; ++p) o0 += q[n++] * W2_0[(e * P2_0 + p) * 128 + c];
        for (int p = 0; p < P1_0; ++p) o0 += q[n++] * W1_0[(e * P1_0 + p) * 128 + c];
        for (int d = 0; d < 3; ++d)
            for (int p = 0; p < P3_1; ++p) o1[d] += q[n++] * W3_1[(e * P3_1 + p) * 128 + c];
        for (int d = 0; d < 3; ++d)
            for (int p = 0; p < P2_1; ++p) o1[d] += q[n++] * W2_1[(e * P2_1 + p) * 128 + c];
        for (int d = 0; d < 3; ++d)
            for (int p = 0; p < P1_1; ++p) o1[d] += q[n++] * W1_1[(e * P1_1 + p) * 128 + c];

        out[b * 512 + c] = o0;                                  // 128x0e block
        #pragma unroll
        for (int d = 0; d < 3; ++d)
            out[b * 512 + 128 + c * 3 + d] = o1[d];             // 128x1o block (mul-major)
    }
}

// ---------------------------------------------------------------------------
// Launch. Input dict order: x, y, U1_0, W1_0, U2_0, W2_0, U3_0, W3_0,
//                           U1_1, W1_1, U2_1, W2_1, U3_1, W3_1
// ---------------------------------------------------------------------------
extern "C" void kernel_launch(void* const* d_in, const int* in_sizes, int n_in,
                              void* d_out, int out_size, void* d_ws, size_t ws_size,
                              hipStream_t stream) {
    const float* x    = (const float*)d_in[0];
    const float* yv   = (const float*)d_in[1];
    const float* U1_0 = (const float*)d_in[2];  const float* W1_0 = (const float*)d_in[3];
    const float* U2_0 = (const float*)d_in[4];  const float* W2_0 = (const float*)d_in[5];
    const float* U3_0 = (const float*)d_in[6];  const float* W3_0 = (const float*)d_in[7];
    const float* U1_1 = (const float*)d_in[8];  const float* W1_1 = (const float*)d_in[9];
    const float* U2_1 = (const float*)d_in[10]; const float* W2_1 = (const float*)d_in[11];
    const float* U3_1 = (const float*)d_in[12]; const float* W3_1 = (const float*)d_in[13];

    // Derive path counts from flat sizes (D0=1, D1=3; n=9).
    int P1_0 = in_sizes[2]  / 9;
    int P2_0 = in_sizes[4]  / 81;
    int P3_0 = in_sizes[6]  / 729;
    int P1_1 = in_sizes[8]  / 27;
    int P2_1 = in_sizes[10] / 243;
    int P3_1 = in_sizes[12] / 2187;

    int B = in_sizes[0] / (128 * 9);   // batch (2048)

    float* Ucat = (float*)d_ws;        // KPAD*NCOLP floats = ~210 KB (L2-resident)

    int packN = KPAD * NCOLP;
    pack_ucat_kernel<<<(packN + 255) / 256, 256, 0, stream>>>(
        U1_0, U2_0, U3_0, U1_1, U2_1, U3_1, Ucat,
        P1_0, P2_0, P3_0, P1_1, P2_1, P3_1);

    // rows = B*128; 16 rows per wave-tile; 8 wave-tiles per 256-thread block -> B blocks.
    symcon_kernel<<<B, 256, 0, stream>>>(
        x, yv, Ucat,
        W1_0, W2_0, W3_0, W1_1, W2_1, W3_1,
        (float*)d_out,
        P1_0, P2_0, P3_0, P1_1, P2_1, P3_1);
}